// AttentionDecoder_2035814499129
// MI455X (gfx1250) — compile-verified
//
#include <hip/hip_runtime.h>
#include <math.h>

// ---------------------------------------------------------------------------
// AttentionDecoder for MI455X (gfx1250, wave32).
// Memory-bound after algebraic reduction: one 205MB streaming pass over
// h_dynamic/h_static doing fused GEMV (compat) + online-softmax-weighted row
// sums. V_WMMA_F32_16X16X4_F32 performs the K-reduction for 16 rows per wave
// per step (A row0 = u vector fed from an interleaved LDS layout with affine
// per-lane addressing; B = h tile in the native 4x16 layout). Next-tile
// global_prefetch_b8 deepens MLP into L2. Small matvecs are tiny kernels;
// sampling reproduces jax.random.categorical via threefry2x32.
// ---------------------------------------------------------------------------

typedef float v2f __attribute__((ext_vector_type(2)));
typedef float v8f __attribute__((ext_vector_type(8)));

#define NROWS 100000
#define HDIM 256
#define NTILES 6250          // NROWS / 16
#define MAIN_BLOCKS 256
#define GUM_BLOCKS 128
#define RECSZ 514            // m, Z, a_s[256], a_d[256]

// workspace layout (float offsets)
#define OF_HC   0
#define OF_Q    256
#define OF_US   512
#define OF_UD   768
#define OF_PART 1024
#define OF_AGV  (OF_PART + MAIN_BLOCKS * RECSZ)   // argmax partial values
#define OF_AGI  (OF_AGV + GUM_BLOCKS)             // argmax partial indices (int)
#define OF_VCN  (OF_AGI + GUM_BLOCKS)             // valid-count partials (int)
#define OF_SCL  (OF_VCN + GUM_BLOCKS)             // per-block softmax scales
#define OF_WSN  (OF_SCL + MAIN_BLOCKS)            // alpha-weighted h_static mean
#define OF_WDN  (OF_WSN + HDIM)                   // alpha-weighted h_dynamic mean
#define OF_CTX  (OF_WDN + HDIM)                   // context
#define OF_HID  (OF_CTX + HDIM)                   // hidden
#define OF_M    (OF_HID + HDIM)
#define OF_Z    (OF_M + 1)
#define OF_V    (OF_Z + 1)                        // valid count (stored as float)
#define OF_CH   (OF_V + 1)                        // choice (stored as float)
// total ~134276 floats (~525 KB) of scratch

// ---------------------------------------------------------------------------
// K1: h_cur = h_s[node] + h_d[node];  q = h_cur @ W_q   (16 blocks x 16 cols)
// ---------------------------------------------------------------------------
__global__ void k_prep_q(const float* __restrict__ hd, const float* __restrict__ hs,
                         const float* __restrict__ Wq, const int* __restrict__ cur,
                         float* __restrict__ ws) {
    __shared__ float hc[HDIM];
    __shared__ float red[256];
    const int tid = threadIdx.x;
    const int node = cur[0];
    hc[tid] = hs[(size_t)node * HDIM + tid] + hd[(size_t)node * HDIM + tid];
    __syncthreads();
    const int jl = tid & 15, c = tid >> 4;
    const int j = blockIdx.x * 16 + jl;
    float s = 0.f;
#pragma unroll
    for (int k = 16 * c; k < 16 * c + 16; ++k) s += hc[k] * Wq[(size_t)k * HDIM + j];
    red[tid] = s;
    __syncthreads();
    for (int off = 128; off >= 16; off >>= 1) {
        if (tid < off) red[tid] += red[tid + off];
        __syncthreads();
    }
    if (tid < 16) ws[OF_Q + blockIdx.x * 16 + tid] = red[tid];
    if (blockIdx.x == 0) ws[OF_HC + tid] = hc[tid];
}

// ---------------------------------------------------------------------------
// K2: u_s[j] = sum_k Wk_s[j,k] q[k]  (Wk = first 256 cols of W_*_kvl [H,3H])
//     32 blocks: b<16 -> static, b>=16 -> dynamic; 16 rows per block
// ---------------------------------------------------------------------------
__global__ void k_prep_u(const float* __restrict__ Ws, const float* __restrict__ Wd,
                         float* __restrict__ ws) {
    __shared__ float q[HDIM];
    __shared__ float red[256];
    const int tid = threadIdx.x;
    const int sel = blockIdx.x >> 4;
    const int jb = (blockIdx.x & 15) * 16;
    const float* W = sel ? Wd : Ws;
    q[tid] = ws[OF_Q + tid];
    __syncthreads();
    const int jl = tid >> 4, c = tid & 15;     // thread group-of-16 spans one row
    float s = 0.f;
#pragma unroll
    for (int k = 16 * c; k < 16 * c + 16; ++k)
        s += W[(size_t)(jb + jl) * 768 + k] * q[k];
    red[tid] = s;
    __syncthreads();
    for (int off = 8; off >= 1; off >>= 1) {
        if (c < off) red[tid] += red[tid + off];
        __syncthreads();
    }
    if (c == 0) ws[OF_US + sel * HDIM + jb + jl] = red[tid];
}

// ---------------------------------------------------------------------------
// K3: MAIN PASS. Per wave: 16-row tiles. WMMA f32 16x16x4 chain computes
// compat for 16 rows; online softmax with wave-uniform rescale; second
// (cache-hot) touch accumulates e-weighted rows. A-operand comes from an
// interleaved LDS image [us0..3 | ud0..3] per k-group so ONE ds_load_b128
// feeds both WMMAs, with affine per-lane addressing (stride 0 for the
// zero-fed lanes). Block partials combined via LDS ds_add_f32.
// ---------------------------------------------------------------------------
__global__ void k_main(const float* __restrict__ hd, const float* __restrict__ hs,
                       const unsigned char* __restrict__ mask,
                       float* __restrict__ ws) {
    __shared__ float lds_u[520];      // 64 groups x 8 floats, zero slot [512..515]
    __shared__ float lds_acc[512];    // block-combined a_s, a_d
    __shared__ float lds_m[8];
    __shared__ float lds_z;

    const int tid = threadIdx.x;
    const int lane = tid & 31;
    const int wave = tid >> 5;

    // interleaved staging: group g = k>>2 holds [us[4g..4g+3], ud[4g..4g+3]]
    {
        const float usv = ws[OF_US + tid];
        const float udv = ws[OF_UD + tid];
        const int g = tid >> 2, e4 = tid & 3;
        lds_u[g * 8 + e4] = usv;
        lds_u[g * 8 + 4 + e4] = udv;
    }
    if (tid < 8) lds_u[512 + tid] = 0.f;   // zero slot (b128-readable)
    lds_acc[tid] = 0.f;
    lds_acc[tid + 256] = 0.f;
    if (tid == 0) lds_z = 0.f;
    __syncthreads();

    // A-operand feed: row M=0 of the 16x4 A matrix lives in lane0 (K=k0,k0+1)
    // and lane16 (K=k0+2,k0+3); every other lane reads the LDS zero slot.
    // Affine: sel lanes advance 8 floats per k-group, others stride 0.
    const bool asel = (lane == 0) || (lane == 16);
    const int aaddr0 = asel ? ((lane == 16) ? 4 : 0) : 512;
    const int astep = asel ? 8 : 0;

    const int gwave = blockIdx.x * 8 + wave;
    const int nwaves = gridDim.x * 8;

    float m = -INFINITY, Z = 0.f;
    float acc_s[8], acc_d[8];
#pragma unroll
    for (int j = 0; j < 8; ++j) { acc_s[j] = 0.f; acc_d[j] = 0.f; }

    for (int t = gwave; t < NTILES; t += nwaves) {
        const int row0 = t * 16;
        // B-operand: lane (n + 16*h) holds B[2h..2h+1][n] = h[row0+n][k0+2h ..]
        const int rB = lane & 15;
        const int kofs = (lane >> 4) * 2;
        const float* bs = hs + (size_t)(row0 + rB) * HDIM + kofs;
        const float* bd = hd + (size_t)(row0 + rB) * HDIM + kofs;

        // prefetch next tile (32KB across both matrices) into L2
        const int t2 = t + nwaves;
        if (t2 < NTILES) {
            const char* ns = (const char*)(hs + (size_t)t2 * 16 * HDIM) + lane * 128;
            const char* nd = (const char*)(hd + (size_t)t2 * 16 * HDIM) + lane * 128;
#pragma unroll
            for (int i = 0; i < 4; ++i) {
                __builtin_prefetch(ns + i * 4096, 0, 1);
                __builtin_prefetch(nd + i * 4096, 0, 1);
            }
        }

        v8f c = {0.f, 0.f, 0.f, 0.f, 0.f, 0.f, 0.f, 0.f};
        int aaddr = aaddr0;
#pragma unroll 8
        for (int k0 = 0; k0 < HDIM; k0 += 4) {
            const float4 au = *(const float4*)&lds_u[aaddr];
            aaddr += astep;
            v2f a; a.x = au.x; a.y = au.y;           // lane0 us[k0,k0+1], lane16 ud[k0,k0+1]
            v2f b; b.x = bs[k0]; b.y = bs[k0 + 1];
            c = __builtin_amdgcn_wmma_f32_16x16x4_f32(false, a, false, b,
                                                      (short)0, c, false, false);
            v2f a2; a2.x = au.z; a2.y = au.w;        // lane0 us[k0+2,k0+3], lane16 ud[k0+2,k0+3]
            v2f b2; b2.x = bd[k0]; b2.y = bd[k0 + 1];
            c = __builtin_amdgcn_wmma_f32_16x16x4_f32(false, a2, false, b2,
                                                      (short)0, c, false, false);
        }
        // D row0 (compat*16) sits in c[0], lanes 0..15
        const int myrow = row0 + (lane & 15);
        float cm;
        if (lane < 16) {
            const float comp = c[0] * 0.0625f;     // 1/sqrt(256)
            cm = (mask[myrow] != 0) ? comp : -1e9f;
        } else {
            cm = -INFINITY;                        // phantom lanes: e == 0 always
        }
        // wave max
        float tmax = cm;
#pragma unroll
        for (int o = 16; o >= 1; o >>= 1) tmax = fmaxf(tmax, __shfl_xor(tmax, o, 32));
        const float m_new = fmaxf(m, tmax);
        const float scale = expf(m - m_new);       // 1.0 when m unchanged
        const float e = expf(cm - m_new);          // 0 for phantom/invalid (once m_new real)
        float esum = e;
#pragma unroll
        for (int o = 16; o >= 1; o >>= 1) esum += __shfl_xor(esum, o, 32);
        Z = Z * scale + esum;
        m = m_new;
#pragma unroll
        for (int j = 0; j < 8; ++j) { acc_s[j] *= scale; acc_d[j] *= scale; }

        // accumulation touch (tile hot in WGP$/L2); lane covers cols 8*lane..+8
        const float* ps = hs + (size_t)row0 * HDIM + lane * 8;
        const float* pd = hd + (size_t)row0 * HDIM + lane * 8;
#pragma unroll 4
        for (int n = 0; n < 16; ++n) {
            const float en = __shfl(e, n, 32);     // wave-uniform broadcast
            if (en != 0.f) {
                const float4 s0 = *(const float4*)(ps + n * HDIM);
                const float4 s1 = *(const float4*)(ps + n * HDIM + 4);
                const float4 d0 = *(const float4*)(pd + n * HDIM);
                const float4 d1 = *(const float4*)(pd + n * HDIM + 4);
                acc_s[0] += en * s0.x; acc_s[1] += en * s0.y;
                acc_s[2] += en * s0.z; acc_s[3] += en * s0.w;
                acc_s[4] += en * s1.x; acc_s[5] += en * s1.y;
                acc_s[6] += en * s1.z; acc_s[7] += en * s1.w;
                acc_d[0] += en * d0.x; acc_d[1] += en * d0.y;
                acc_d[2] += en * d0.z; acc_d[3] += en * d0.w;
                acc_d[4] += en * d1.x; acc_d[5] += en * d1.y;
                acc_d[6] += en * d1.z; acc_d[7] += en * d1.w;
            }
        }
    }

    // block combine (8 waves) via LDS
    if (lane == 0) lds_m[wave] = m;
    __syncthreads();
    float bm = lds_m[0];
#pragma unroll
    for (int w = 1; w < 8; ++w) bm = fmaxf(bm, lds_m[w]);
    const float bscale = expf(m - bm);
#pragma unroll
    for (int j = 0; j < 8; ++j) {
        atomicAdd(&lds_acc[lane * 8 + j], acc_s[j] * bscale);
        atomicAdd(&lds_acc[256 + lane * 8 + j], acc_d[j] * bscale);
    }
    if (lane == 0) atomicAdd(&lds_z, Z * bscale);
    __syncthreads();

    float* rec = ws + OF_PART + (size_t)blockIdx.x * RECSZ;
    if (tid == 0) { rec[0] = bm; rec[1] = lds_z; }
    rec[2 + tid] = lds_acc[tid];
    rec[2 + 256 + tid] = lds_acc[256 + tid];
}

// ---------------------------------------------------------------------------
// K4: threefry2x32 Gumbel argmax + valid count (reproduces
// jax.random.categorical(key(1), logits_all)).  key(1) -> (k0,k1) = (0,1).
// counts iota(100000) split: pair p -> (p, p+50000), out0 -> elt p, out1 -> p+5e4
// ---------------------------------------------------------------------------
__device__ inline void threefry2x32(unsigned k0, unsigned k1,
                                    unsigned x0, unsigned x1,
                                    unsigned& o0, unsigned& o1) {
    const unsigned k2 = k0 ^ k1 ^ 0x1BD11BDAu;
    const unsigned ks[3] = {k0, k1, k2};
    x0 += k0; x1 += k1;
    const int rot[2][4] = {{13, 15, 26, 6}, {17, 29, 16, 24}};
#pragma unroll
    for (int i = 0; i < 5; ++i) {
        const int* r = rot[i & 1];
#pragma unroll
        for (int j = 0; j < 4; ++j) {
            x0 += x1;
            x1 = (x1 << r[j]) | (x1 >> (32 - r[j]));
            x1 ^= x0;
        }
        x0 += ks[(i + 1) % 3];
        x1 += ks[(i + 2) % 3] + (unsigned)(i + 1);
    }
    o0 = x0; o1 = x1;
}

__device__ inline float gumbel_from_bits(unsigned b) {
    const float TINY = 1.17549435e-38f;
    float f = __uint_as_float(0x3F800000u | (b >> 9)) - 1.0f;   // [0,1)
    float u = fmaxf(TINY, f * (1.0f - TINY) + TINY);
    return -logf(-logf(u));
}

__global__ void k_gumbel(const unsigned char* __restrict__ mask,
                         float* __restrict__ ws) {
    __shared__ float bv[256];
    __shared__ int bi[256];
    __shared__ int bc[256];
    const int tid = threadIdx.x;
    const int gid = blockIdx.x * blockDim.x + tid;
    const int stride = gridDim.x * blockDim.x;
    const int HALF = NROWS / 2;

    float bestv = -INFINITY;
    int besti = 0x7FFFFFFF;
    int cnt = 0;
    for (int p = gid; p < HALF; p += stride) {
        unsigned o0, o1;
        threefry2x32(0u, 1u, (unsigned)p, (unsigned)(p + HALF), o0, o1);
        const int v0 = mask[p] != 0, v1 = mask[p + HALF] != 0;
        const float key0 = (v0 ? 0.f : -1e9f) + gumbel_from_bits(o0);
        const float key1 = (v1 ? 0.f : -1e9f) + gumbel_from_bits(o1);
        cnt += v0 + v1;
        if (key0 > bestv || (key0 == bestv && p < besti)) { bestv = key0; besti = p; }
        if (key1 > bestv || (key1 == bestv && p + HALF < besti)) { bestv = key1; besti = p + HALF; }
    }
    bv[tid] = bestv; bi[tid] = besti; bc[tid] = cnt;
    __syncthreads();
    for (int off = 128; off >= 1; off >>= 1) {
        if (tid < off) {
            if (bv[tid + off] > bv[tid] ||
                (bv[tid + off] == bv[tid] && bi[tid + off] < bi[tid])) {
                bv[tid] = bv[tid + off]; bi[tid] = bi[tid + off];
            }
            bc[tid] += bc[tid + off];
        }
        __syncthreads();
    }
    if (tid == 0) {
        ws[OF_AGV + blockIdx.x] = bv[0];
        ((int*)ws)[OF_AGI + blockIdx.x] = bi[0];
        ((int*)ws)[OF_VCN + blockIdx.x] = bc[0];
    }
}

// ---------------------------------------------------------------------------
// K5: single block: global M, scales[], Z; argmax & valid-count finalization
// ---------------------------------------------------------------------------
__global__ void k_reduce1(float* __restrict__ ws) {
    __shared__ float r[256];
    __shared__ int ri[256];
    const int tid = threadIdx.x;

    const float mb = ws[OF_PART + (size_t)tid * RECSZ];
    r[tid] = mb;
    __syncthreads();
    for (int off = 128; off >= 1; off >>= 1) {
        if (tid < off) r[tid] = fmaxf(r[tid], r[tid + off]);
        __syncthreads();
    }
    const float M = r[0];
    __syncthreads();
    const float sc = expf(mb - M);
    ws[OF_SCL + tid] = sc;
    r[tid] = ws[OF_PART + (size_t)tid * RECSZ + 1] * sc;
    __syncthreads();
    for (int off = 128; off >= 1; off >>= 1) {
        if (tid < off) r[tid] += r[tid + off];
        __syncthreads();
    }
    if (tid == 0) { ws[OF_Z] = r[0]; ws[OF_M] = M; }
    __syncthreads();

    // argmax over GUM_BLOCKS partials + valid count
    float av = (tid < GUM_BLOCKS) ? ws[OF_AGV + tid] : -INFINITY;
    int ai = (tid < GUM_BLOCKS) ? ((int*)ws)[OF_AGI + tid] : 0x7FFFFFFF;
    int vc = (tid < GUM_BLOCKS) ? ((int*)ws)[OF_VCN + tid] : 0;
    r[tid] = av; ri[tid] = ai;
    __syncthreads();
    for (int off = 128; off >= 1; off >>= 1) {
        if (tid < off) {
            if (r[tid + off] > r[tid] ||
                (r[tid + off] == r[tid] && ri[tid + off] < ri[tid])) {
                r[tid] = r[tid + off]; ri[tid] = ri[tid + off];
            }
        }
        __syncthreads();
    }
    if (tid == 0) ws[OF_CH] = (float)ri[0];
    __syncthreads();
    r[tid] = (float)vc;
    __syncthreads();
    for (int off = 128; off >= 1; off >>= 1) {
        if (tid < off) r[tid] += r[tid + off];
        __syncthreads();
    }
    if (tid == 0) ws[OF_V] = r[0];
}

// ---------------------------------------------------------------------------
// K6: alpha-weighted h means: wsn/wdn[j] = (sum_b a_b[j]*scale_b) / Z
// ---------------------------------------------------------------------------
__global__ void k_wsum(float* __restrict__ ws) {
    __shared__ float sl[MAIN_BLOCKS];
    __shared__ float red[256];
    const int tid = threadIdx.x;
    sl[tid] = ws[OF_SCL + tid];
    __syncthreads();
    const int sel = blockIdx.x >> 4;
    const int jb = (blockIdx.x & 15) * 16;
    const int jl = tid & 15, c = tid >> 4;
    const int j = jb + jl;
    float s = 0.f;
    for (int r0 = c * 16; r0 < c * 16 + 16; ++r0)
        s += ws[OF_PART + (size_t)r0 * RECSZ + 2 + sel * HDIM + j] * sl[r0];
    red[tid] = s;
    __syncthreads();
    for (int off = 128; off >= 16; off >>= 1) {
        if (tid < off) red[tid] += red[tid + off];
        __syncthreads();
    }
    if (tid < 16) {
        const float invZ = 1.0f / ws[OF_Z];
        ws[OF_WSN + sel * HDIM + jb + tid] = red[tid] * invZ;
    }
}

// ---------------------------------------------------------------------------
// K7: context[j] = sum_k wsn[k]*Wv_s[k,j] + wdn[k]*Wv_d[k,j]  (Wv = cols 256..511)
// ---------------------------------------------------------------------------
__global__ void k_context(const float* __restrict__ Ws, const float* __restrict__ Wd,
                          float* __restrict__ ws) {
    __shared__ float wv[2 * HDIM];
    __shared__ float red[256];
    const int tid = threadIdx.x;
    wv[tid] = ws[OF_WSN + tid];
    wv[HDIM + tid] = ws[OF_WDN + tid];
    __syncthreads();
    const int jl = tid & 15, c = tid >> 4;
    const int j = blockIdx.x * 16 + jl;
    float s = 0.f;
    for (int k = 16 * c; k < 16 * c + 16; ++k) {
        s += wv[k] * Ws[(size_t)k * 768 + 256 + j];
        s += wv[HDIM + k] * Wd[(size_t)k * 768 + 256 + j];
    }
    red[tid] = s;
    __syncthreads();
    for (int off = 128; off >= 16; off >>= 1) {
        if (tid < off) red[tid] += red[tid + off];
        __syncthreads();
    }
    if (tid < 16) ws[OF_CTX + blockIdx.x * 16 + tid] = red[tid];
}

// ---------------------------------------------------------------------------
// K8: hidden = relu([h_cur, context] @ W1 + b1)
// ---------------------------------------------------------------------------
__global__ void k_mlp(const float* __restrict__ W1, const float* __restrict__ b1,
                      float* __restrict__ ws) {
    __shared__ float fuse[2 * HDIM];
    __shared__ float red[256];
    const int tid = threadIdx.x;
    fuse[tid] = ws[OF_HC + tid];
    fuse[HDIM + tid] = ws[OF_CTX + tid];
    __syncthreads();
    const int jl = tid & 15, c = tid >> 4;
    const int j = blockIdx.x * 16 + jl;
    float s = 0.f;
    for (int t = 32 * c; t < 32 * c + 32; ++t)
        s += fuse[t] * W1[(size_t)t * HDIM + j];
    red[tid] = s;
    __syncthreads();
    for (int off = 128; off >= 16; off >>= 1) {
        if (tid < off) red[tid] += red[tid + off];
        __syncthreads();
    }
    if (tid < 16) {
        const float h = red[tid] + b1[j];
        ws[OF_HID + j] = h > 0.f ? h : 0.f;
    }
}

// ---------------------------------------------------------------------------
// K9: logit + faithful log_softmax path + outputs (choice, log_prob)
// ---------------------------------------------------------------------------
__global__ void k_final(const float* __restrict__ W2, const float* __restrict__ b2,
                        const float* __restrict__ ws, float* __restrict__ out) {
    __shared__ float red[256];
    const int tid = threadIdx.x;
    red[tid] = ws[OF_HID + tid] * W2[tid];
    __syncthreads();
    for (int off = 128; off >= 1; off >>= 1) {
        if (tid < off) red[tid] += red[tid + off];
        __syncthreads();
    }
    if (tid == 0) {
        const float logit = red[0] + b2[0];
        const float Vf = ws[OF_V];
        float lp;
        if (Vf > 0.f) {
            // logits_all: V entries == logit, rest -1e9.  max = logit,
            // sum(exp(x-max)) == Vf exactly -> lp = logit - (logit + log Vf)
            lp = logit - (logit + logf(Vf));
        } else {
            lp = -1e9f - (-1e9f + logf((float)NROWS));
        }
        out[0] = ws[OF_CH];
        out[1] = lp;
    }
}

// ---------------------------------------------------------------------------
extern "C" void kernel_launch(void* const* d_in, const int* in_sizes, int n_in,
                              void* d_out, int out_size, void* d_ws, size_t ws_size,
                              hipStream_t stream) {
    const float* hd = (const float*)d_in[0];
    const float* hs = (const float*)d_in[1];
    const float* Ws = (const float*)d_in[2];
    const float* Wd = (const float*)d_in[3];
    const float* Wq = (const float*)d_in[4];
    const float* W1 = (const float*)d_in[5];
    const float* b1 = (const float*)d_in[6];
    const float* W2 = (const float*)d_in[7];
    const float* b2 = (const float*)d_in[8];
    const unsigned char* mask = (const unsigned char*)d_in[9];
    const int* cur = (const int*)d_in[10];
    float* ws = (float*)d_ws;
    float* out = (float*)d_out;

    k_prep_q<<<16, 256, 0, stream>>>(hd, hs, Wq, cur, ws);
    k_prep_u<<<32, 256, 0, stream>>>(Ws, Wd, ws);
    k_main<<<MAIN_BLOCKS, 256, 0, stream>>>(hd, hs, mask, ws);
    k_gumbel<<<GUM_BLOCKS, 256, 0, stream>>>(mask, ws);
    k_reduce1<<<1, 256, 0, stream>>>(ws);
    k_wsum<<<32, 256, 0, stream>>>(ws);
    k_context<<<16, 256, 0, stream>>>(Ws, Wd, ws);
    k_mlp<<<16, 256, 0, stream>>>(W1, b1, ws);
    k_final<<<1, 256, 0, stream>>>(W2, b2, ws, out);
}